// MultiTimeAttention_10685878633066
// MI455X (gfx1250) — compile-verified
//
#include <hip/hip_runtime.h>
#include <math.h>

// Problem constants (from reference)
#define BB   16
#define HH   4
#define HID  128
#define DKK  32
#define DD   64
#define OUTD 128
#define SQL  128
#define SKL  256

typedef float v2f __attribute__((ext_vector_type(2)));
typedef float v8f __attribute__((ext_vector_type(8)));

__device__ __forceinline__ v8f wmma_f32(v2f a, v2f b, v8f c) {
    // D = A(16x4) * B(4x16) + C, full f32. 8-arg form:
    // (neg_a, A, neg_b, B, c_mod, C, reuse_a, reuse_b)
    return __builtin_amdgcn_wmma_f32_16x16x4_f32(
        false, a, false, b, (short)0, c, false, false);
}

__device__ __forceinline__ v2f ld2(const float* p) { return *(const v2f*)p; }

// ---------------------------------------------------------------------------
// Kernel 1/2: input projection  out[b,h,s,dk] = (in[b,s,:] @ W^T + bias)[h*DK+dk]
// One wave = one 16x16 tile of the (B*S, HID) output. K = HID.
// ---------------------------------------------------------------------------
__global__ void proj_kernel(const float* __restrict__ in, const float* __restrict__ W,
                            const float* __restrict__ bias, float* __restrict__ outp,
                            int S) {
    const int lane = threadIdx.x & 31;
    const int hl   = lane >> 4;     // half-wave select (K pair)
    const int l16  = lane & 15;
    const int tile = blockIdx.x * (blockDim.x >> 5) + (threadIdx.x >> 5);
    const int tn = tile & 7;        // HID/16 = 8 column tiles
    const int tm = tile >> 3;
    const int m0 = tm * 16;
    const int n0 = tn * 16;

    const float* arow = in + (size_t)(m0 + l16) * HID;        // A[m][k]
    const float* brow = W  + (size_t)(n0 + l16) * HID;        // B[k][n] = W[n][k]
    v8f c = {};
    for (int kk = 0; kk < HID; kk += 4) {
        c = wmma_f32(ld2(arow + kk + 2 * hl), ld2(brow + kk + 2 * hl), c);
    }
    const int   n  = n0 + l16;
    const float bv = bias[n];
    const int   h  = n >> 5;        // DK = 32
    const int   dk = n & 31;
    for (int j = 0; j < 8; ++j) {
        int grow = m0 + j + 8 * hl;
        int b    = grow / S;
        int s    = grow - b * S;
        outp[(((size_t)b * HH + h) * S + s) * DKK + dk] = c[j] + bv;
    }
}

// ---------------------------------------------------------------------------
// Kernel 3: raw scores  S[bh,q,k] = (Q_h K_h^T)/sqrt(DK).  K = 32.
// ---------------------------------------------------------------------------
__global__ void scores_kernel(const float* __restrict__ Qp, const float* __restrict__ Kp,
                              float* __restrict__ E) {
    const int lane = threadIdx.x & 31, hl = lane >> 4, l16 = lane & 15;
    const int tile = blockIdx.x * (blockDim.x >> 5) + (threadIdx.x >> 5);
    const int bh = tile >> 7;       // (SQ/16)*(SK/16) = 128 tiles per bh
    const int t  = tile & 127;
    const int tm = t >> 4, tn = t & 15;

    const float* arow = Qp + ((size_t)bh * SQL + tm * 16 + l16) * DKK;
    const float* brow = Kp + ((size_t)bh * SKL + tn * 16 + l16) * DKK; // B[k][n]=Kp[n][k]
    v8f c = {};
    for (int kk = 0; kk < DKK; kk += 4) {
        c = wmma_f32(ld2(arow + kk + 2 * hl), ld2(brow + kk + 2 * hl), c);
    }
    const float scale = 0.17677669529663687f;   // 1/sqrt(32)
    float* orow = E + ((size_t)bh * SQL + tm * 16) * SKL + tn * 16 + l16;
    for (int j = 0; j < 8; ++j)
        orow[(size_t)(j + 8 * hl) * SKL] = c[j] * scale;
}

// ---------------------------------------------------------------------------
// Kernel 4: in-place E = exp(S - rowmax).  One wave32 per 256-wide row.
// ---------------------------------------------------------------------------
__global__ void expnorm_kernel(float* __restrict__ E) {
    const int lane = threadIdx.x & 31;
    const int row  = blockIdx.x * (blockDim.x >> 5) + (threadIdx.x >> 5);
    float* r = E + (size_t)row * SKL;
    float v[8];
    float m = -1e30f;
    for (int i = 0; i < 8; ++i) { v[i] = r[lane + 32 * i]; m = fmaxf(m, v[i]); }
    for (int off = 16; off > 0; off >>= 1)
        m = fmaxf(m, __shfl_xor(m, off, 32));
    for (int i = 0; i < 8; ++i)
        r[lane + 32 * i] = __expf(v[i] - m);
}

// ---------------------------------------------------------------------------
// Kernel 5: fused masked attention reduce.
//   num = E @ (mask*value),  den = E @ mask,  X = num/den.
// Two WMMA accumulators share one A operand per K-step. K = 256.
// ---------------------------------------------------------------------------
__global__ void attn_kernel(const float* __restrict__ E, const float* __restrict__ value,
                            const int* __restrict__ mask, float* __restrict__ X) {
    const int lane = threadIdx.x & 31, hl = lane >> 4, l16 = lane & 15;
    const int tile = blockIdx.x * (blockDim.x >> 5) + (threadIdx.x >> 5);
    const int bh = tile >> 5;       // (SQ/16)*(D/16) = 32 tiles per bh
    const int t  = tile & 31;
    const int tm = t >> 2, tn = t & 3;
    const int b  = bh >> 2;         // H = 4

    const float* arow = E + ((size_t)bh * SQL + tm * 16 + l16) * SKL;
    const int    d    = tn * 16 + l16;
    const float* vcol = value + (size_t)b * SKL * DD + d;
    const int*   mcol = mask  + (size_t)b * SKL * DD + d;

    v8f cn = {}, cd = {};
    for (int kk = 0; kk < SKL; kk += 4) {
        int k0 = kk + 2 * hl;
        v2f a  = ld2(arow + k0);
        int   mi0 = mcol[(size_t)k0 * DD];
        int   mi1 = mcol[(size_t)(k0 + 1) * DD];
        float v0  = vcol[(size_t)k0 * DD];
        float v1  = vcol[(size_t)(k0 + 1) * DD];
        v2f bn, bd;
        bn.x = mi0 ? v0 : 0.0f;    bn.y = mi1 ? v1 : 0.0f;
        bd.x = mi0 ? 1.0f : 0.0f;  bd.y = mi1 ? 1.0f : 0.0f;
        cn = wmma_f32(a, bn, cn);
        cd = wmma_f32(a, bd, cd);
    }
    float* orow = X + ((size_t)bh * SQL + tm * 16) * DD + d;
    for (int j = 0; j < 8; ++j)
        orow[(size_t)(j + 8 * hl) * DD] = cn[j] / cd[j];
}

// ---------------------------------------------------------------------------
// Kernel 6: output projection  out = Xcat(2048x256) @ Wo^T + bo.  K = 256.
// Xcat column c = h*64+d is gathered from head-split X layout.
// ---------------------------------------------------------------------------
__global__ void outproj_kernel(const float* __restrict__ X, const float* __restrict__ Wo,
                               const float* __restrict__ bo, float* __restrict__ out) {
    const int lane = threadIdx.x & 31, hl = lane >> 4, l16 = lane & 15;
    const int tile = blockIdx.x * (blockDim.x >> 5) + (threadIdx.x >> 5);
    const int tn = tile & 7;        // OUT/16 = 8
    const int tm = tile >> 3;
    const int m0 = tm * 16, n0 = tn * 16;

    const int grow = m0 + l16;      // A-matrix row for this lane
    const int b    = grow >> 7;     // SQ = 128
    const int s    = grow & 127;
    const float* brow = Wo + (size_t)(n0 + l16) * (HH * DD); // B[k][n]=Wo[n][k]

    v8f c = {};
    for (int kk = 0; kk < HH * DD; kk += 4) {
        int k0 = kk + 2 * hl;
        int h  = k0 >> 6, dd = k0 & 63;   // dd even, stays inside one head block
        v2f a = ld2(X + (((size_t)b * HH + h) * SQL + s) * DD + dd);
        v2f w = ld2(brow + k0);
        c = wmma_f32(a, w, c);
    }
    const float bv = bo[n0 + l16];
    for (int j = 0; j < 8; ++j) {
        int r = m0 + j + 8 * hl;
        out[(size_t)r * OUTD + n0 + l16] = c[j] + bv;
    }
}

// ---------------------------------------------------------------------------
extern "C" void kernel_launch(void* const* d_in, const int* in_sizes, int n_in,
                              void* d_out, int out_size, void* d_ws, size_t ws_size,
                              hipStream_t stream) {
    const float* query = (const float*)d_in[0];
    const float* key   = (const float*)d_in[1];
    const float* value = (const float*)d_in[2];
    const int*   mask  = (const int*)  d_in[3];
    const float* Wq    = (const float*)d_in[4];
    const float* bq    = (const float*)d_in[5];
    const float* Wk    = (const float*)d_in[6];
    const float* bk    = (const float*)d_in[7];
    const float* Wo    = (const float*)d_in[8];
    const float* bo    = (const float*)d_in[9];
    float* out = (float*)d_out;

    // workspace layout (floats): Qp | Kp | E | X  (~13.6 MB total)
    float* Qp = (float*)d_ws;                                  // B*H*SQ*DK
    float* Kp = Qp + (size_t)BB * HH * SQL * DKK;              // B*H*SK*DK
    float* E  = Kp + (size_t)BB * HH * SKL * DKK;              // B*H*SQ*SK
    float* X  = E  + (size_t)BB * HH * SQL * SKL;              // B*H*SQ*D

    dim3 blk(256);  // 8 wave32s per block
    // grids: total 16x16 tiles / 8 waves per block (all divide exactly)
    proj_kernel   <<<(BB * SQL / 16) * (HID / 16) / 8,        blk, 0, stream>>>(query, Wq, bq, Qp, SQL);
    proj_kernel   <<<(BB * SKL / 16) * (HID / 16) / 8,        blk, 0, stream>>>(key,   Wk, bk, Kp, SKL);
    scores_kernel <<<BB * HH * (SQL / 16) * (SKL / 16) / 8,   blk, 0, stream>>>(Qp, Kp, E);
    expnorm_kernel<<<BB * HH * SQL / 8,                       blk, 0, stream>>>(E);
    attn_kernel   <<<BB * HH * (SQL / 16) * (DD / 16) / 8,    blk, 0, stream>>>(E, value, mask, X);
    outproj_kernel<<<(BB * SQL / 16) * (OUTD / 16) / 8,       blk, 0, stream>>>(X, Wo, bo, out);
}